// UpperBandEntropyLoss16_23295902614020
// MI455X (gfx1250) — compile-verified
//
#include <hip/hip_runtime.h>
#include <math.h>

typedef __attribute__((ext_vector_type(2))) float v2f;
typedef __attribute__((ext_vector_type(8))) float v8f;

#define PI_OVER_32 0.09817477042468103f
#define INV_SQRT2 0.70710678118654752f
#define SQRT_DCT_SCALE 0.42044820762685725f /* (1/32)^(1/4) = sqrt(1/sqrt(32)) */

// Wave32 xor-butterfly sum using DS_SWIZZLE_B32 group-of-32 xor mode:
// offset = (xor_mask << 10) | 0x1f (and_mask), per CDNA5 ISA §11.
__device__ __forceinline__ float waveSum32(float v) {
  float r = v;
  r += __builtin_bit_cast(
      float, __builtin_amdgcn_ds_swizzle(__builtin_bit_cast(int, r), 0x401f)); // SWAPX16
  r += __builtin_bit_cast(
      float, __builtin_amdgcn_ds_swizzle(__builtin_bit_cast(int, r), 0x201f)); // SWAPX8
  r += __builtin_bit_cast(
      float, __builtin_amdgcn_ds_swizzle(__builtin_bit_cast(int, r), 0x101f)); // SWAPX4
  r += __builtin_bit_cast(
      float, __builtin_amdgcn_ds_swizzle(__builtin_bit_cast(int, r), 0x081f)); // SWAPX2
  r += __builtin_bit_cast(
      float, __builtin_amdgcn_ds_swizzle(__builtin_bit_cast(int, r), 0x041f)); // SWAPX1
  return r;
}

// One-time (per launch) scaled DCT basis:
//   htab[k*16+n] = cos((2k+1)*n*pi/32) * (n==0 ? 1/sqrt2 : 1) * (1/32)^(1/4)
// Column scaling + sqrt(SCALE) folded in so that coeff = H'^T @ X @ H'
// equals SCALE * NMAT * (H^T @ X @ H) with no per-block post-scale.
__global__ __launch_bounds__(256) void init_harm_kernel(float* __restrict__ htab) {
  const int t = threadIdx.x;
  const int k = t >> 4;
  const int n = t & 15;
  const float cs = cosf((float)((2 * k + 1) * n) * PI_OVER_32);
  htab[t] = cs * ((n == 0) ? INV_SQRT2 : 1.0f) * SQRT_DCT_SCALE;
}

// One wave32 per 16x16 block. 8 waves (8 blocks) per workgroup.
__global__ __launch_bounds__(256) void dct_entropy_kernel(
    const float* __restrict__ x, const float* __restrict__ htab,
    float* __restrict__ ws) {
  __shared__ float smem[8][256];  // per-wave transpose staging tile
  __shared__ float partial[8];

  const int tid = threadIdx.x;
  const int wave = tid >> 5;
  const int lane = tid & 31;
  const int n = lane & 15;   // column within tile
  const int h = lane >> 4;   // lane half
  const int halfK = h << 1;  // 0 / 2

  // block decomposition: 4096 blocks per 1024x1024 image, 64 per block-row
  const int blk = blockIdx.x * 8 + wave;
  const int img = blk >> 12;
  const int rem = blk & 4095;
  const int by = rem >> 6;
  const int bx = rem & 63;
  const float* base =
      x + (size_t)img * (1024u * 1024u) + (size_t)by * (16u * 1024u) + (size_t)bx * 16u;

  // B-layout chunks of X: element v of chunk kc holds X[4kc + halfK + v][n]
  v2f xb[4];
#pragma unroll
  for (int kc = 0; kc < 4; ++kc) {
    const int k0 = 4 * kc + halfK;
    xb[kc].x = base[(size_t)k0 * 1024u + (size_t)n];
    xb[kc].y = base[(size_t)(k0 + 1) * 1024u + (size_t)n];
  }

  // H' chunks straight from the L2-hot global table (one vaddr, 8 imm-offset
  // loads): element j of chunk kc holds H'[4kc+halfK+j][n]. Simultaneously
  // the A-layout of H'^T (matmul 1) and the B-layout of H' (matmul 2).
  const float* hbase = htab + (size_t)(halfK * 16 + n);
  v2f hv[4];
#pragma unroll
  for (int kc = 0; kc < 4; ++kc) {
    hv[kc].x = hbase[kc * 64];
    hv[kc].y = hbase[kc * 64 + 16];
  }

  // ---- matmul 1: M1 = H'^T @ X ----
  v8f m1 = {0.f, 0.f, 0.f, 0.f, 0.f, 0.f, 0.f, 0.f};
#pragma unroll
  for (int kc = 0; kc < 4; ++kc) {
    m1 = __builtin_amdgcn_wmma_f32_16x16x4_f32(false, hv[kc], false, xb[kc],
                                               (short)0, m1, false, false);
  }

  // ---- re-layout M1 (C-format -> A-format) through per-wave LDS tile.
  // LDS ops are in-order within a wave; wave_barrier pins compiler ordering.
#pragma unroll
  for (int v = 0; v < 8; ++v)
    smem[wave][(v + (h << 3)) * 16 + n] = m1[v];
  __builtin_amdgcn_wave_barrier();

  v2f a2[4];
#pragma unroll
  for (int kc = 0; kc < 4; ++kc) {
    const float* p = &smem[wave][n * 16 + 4 * kc + halfK];
    a2[kc].x = p[0];
    a2[kc].y = p[1];
  }

  // ---- matmul 2: coeff = M1 @ H' (already carries SCALE * NMAT) ----
  v8f c2 = {0.f, 0.f, 0.f, 0.f, 0.f, 0.f, 0.f, 0.f};
#pragma unroll
  for (int kc = 0; kc < 4; ++kc) {
    c2 = __builtin_amdgcn_wmma_f32_16x16x4_f32(false, a2[kc], false, hv[kc],
                                               (short)0, c2, false, false);
  }

  // ---- upper-band membership (reductions are permutation-invariant, so the
  // zigzag order is irrelevant): kept iff m+n>=16 or (m+n==15 and m>=8),
  // m = v + 8h  =>  v >= thr with thr = 16 - 9h - n. Exactly 128 kept.
  // Apply the mask ONCE; downstream, masked-out entries behave identically to
  // the reference's thresholded entries (a = eps), a ~1e-9-relative perturbation.
  const int thr = 16 - 9 * h - n;
  float cm[8];
#pragma unroll
  for (int v = 0; v < 8; ++v) cm[v] = (v >= thr) ? c2[v] : 0.0f;

  // mean over the 128 kept coefficients (masked entries add exactly 0)
  const float s1 = ((cm[0] + cm[1]) + (cm[2] + cm[3])) +
                   ((cm[4] + cm[5]) + (cm[6] + cm[7]));
  const float mean = waveSum32(s1) * (1.0f / 128.0f);

  // threshold-below-mean, abs + eps, L2 norm (no re-masking needed:
  // out-of-band cm==0 yields a==eps via either branch)
  float a[8];
  float s2 = 0.f;
#pragma unroll
  for (int v = 0; v < 8; ++v) {
    a[v] = (cm[v] < mean) ? 1e-12f : fabsf(cm[v]) + 1e-12f;
    s2 += a[v] * a[v];
  }
  const float invD = 1.0f / fmaxf(sqrtf(waveSum32(s2)), 1e-12f);

  // entropy: p >= eps/D > 0 for all 8 entries -> no mask, no NaN.
  // __log2f -> single v_log_f32.
  float s3 = 0.f;
#pragma unroll
  for (int v = 0; v < 8; ++v) {
    const float p = a[v] * invD;
    s3 += p * __log2f(p);
  }
  const float ent = -waveSum32(s3);

  if (lane == 0) partial[wave] = ent;
  __syncthreads();
  if (tid == 0) {
    float t = 0.f;
#pragma unroll
    for (int i = 0; i < 8; ++i) t += partial[i];
    ws[blockIdx.x] = t;
  }
}

// deterministic final reduction (also initializes d_out, which is poisoned)
__global__ __launch_bounds__(256) void reduce_ws_kernel(
    const float* __restrict__ ws, float* __restrict__ out, int n, float scale) {
  __shared__ float s[256];
  const int t = threadIdx.x;
  float acc = 0.f;
  for (int i = t; i < n; i += 256) acc += ws[i];
  s[t] = acc;
  __syncthreads();
  if (t == 0) {
    float tot = 0.f;
    for (int i = 0; i < 256; ++i) tot += s[i];
    out[0] = tot * scale;
  }
}

extern "C" void kernel_launch(void* const* d_in, const int* in_sizes, int n_in,
                              void* d_out, int out_size, void* d_ws, size_t ws_size,
                              hipStream_t stream) {
  const float* x = (const float*)d_in[0];
  float* out = (float*)d_out;
  float* wsf = (float*)d_ws;

  float* htab = wsf;            // [0..255]   scaled DCT basis table
  float* partials = wsf + 256;  // [256..)    per-workgroup partial sums

  const int total = in_sizes[0];          // 8*3*1024*1024 = 25165824
  const int nBlocks16 = total / 256;      // 98304 16x16 blocks
  const int nWG = nBlocks16 / 8;          // 12288 workgroups, 8 blocks each

  init_harm_kernel<<<1, 256, 0, stream>>>(htab);
  dct_entropy_kernel<<<nWG, 256, 0, stream>>>(x, htab, partials);
  reduce_ws_kernel<<<1, 256, 0, stream>>>(partials, out, nWG, 1.0f / (float)nBlocks16);
}